// CrossAttentionLayer_23356032155662
// MI455X (gfx1250) — compile-verified
//
#include <hip/hip_runtime.h>
#include <hip/hip_bf16.h>

typedef __attribute__((ext_vector_type(2))) float v2f;
typedef __attribute__((ext_vector_type(8))) float v8f;

#define NH   8
#define HD   32
#define PDIM 1024
#define QDIM 8192
#define CDIM 256

__device__ __forceinline__ v8f wmma4(v2f a, v2f b, v8f c) {
  // D = A(16x4) * B(4x16) + C(16x16), fp32
  return __builtin_amdgcn_wmma_f32_16x16x4_f32(
      /*neg_a=*/false, a, /*neg_b=*/false, b,
      /*c_mod=*/(short)0, c, /*reuse_a=*/false, /*reuse_b=*/false);
}

// ---------------------------------------------------------------------------
// Projection GEMM: Out[o][s] = sum_c W[o][c] * X[c][s] + bias[o]
// X layout: [b][C][S] (channel-major, spatial contiguous) -- matches the
// query/key/value tensors.  Output stored head-split: [b][n][s][hd].
// Each wave computes a 16(o) x 64(s) tile: A fragment (W) reused 4x.
// ---------------------------------------------------------------------------
__global__ void proj_cs_kernel(const float* __restrict__ W,
                               const float* __restrict__ bias,
                               const float* __restrict__ X,
                               float* __restrict__ Out, int S) {
  const int lane = threadIdx.x & 31;
  const int lo = lane & 15, hi = lane >> 4;
  const int wave = threadIdx.x >> 5;
  const int b  = blockIdx.z;
  const int m0 = blockIdx.y * 16;
  const int s0 = (blockIdx.x * 4 + wave) * 64;
  const float* Xb = X + (size_t)b * CDIM * S;

  v8f acc0 = {}, acc1 = {}, acc2 = {}, acc3 = {};
  #pragma unroll 4
  for (int k = 0; k < CDIM; k += 4) {
    // A frag: A[m=lo, kk=v+2*hi] = W[(m0+lo)][k+2hi .. +1]  (8B-aligned)
    v2f A = *(const v2f*)(W + (size_t)(m0 + lo) * CDIM + k + 2 * hi);
    const float* xr0 = Xb + (size_t)(k + 2 * hi) * S + s0 + lo;
    const float* xr1 = xr0 + S;
    v2f B0 = { xr0[0],  xr1[0]  };
    v2f B1 = { xr0[16], xr1[16] };
    v2f B2 = { xr0[32], xr1[32] };
    v2f B3 = { xr0[48], xr1[48] };
    acc0 = wmma4(A, B0, acc0);
    acc1 = wmma4(A, B1, acc1);
    acc2 = wmma4(A, B2, acc2);
    acc3 = wmma4(A, B3, acc3);
  }
  #pragma unroll
  for (int r = 0; r < 8; ++r) {
    int o = m0 + r + 8 * hi;           // C layout: row M = r (+8 for hi lanes)
    float bb = bias[o];
    int n = o >> 5, d = o & 31;
    size_t rowbase = ((size_t)b * NH + n) * S;
    Out[(rowbase + s0 +  0 + lo) * HD + d] = acc0[r] + bb;
    Out[(rowbase + s0 + 16 + lo) * HD + d] = acc1[r] + bb;
    Out[(rowbase + s0 + 32 + lo) * HD + d] = acc2[r] + bb;
    Out[(rowbase + s0 + 48 + lo) * HD + d] = acc3[r] + bb;
  }
}

// ---------------------------------------------------------------------------
// Fused attention: energy (transposed: q rows x p cols), online softmax over
// q, recompute pass writes normalized A and accumulates A*V via an LDS
// transpose bounce.  One wave per (b, head, 16-wide p tile).
// ---------------------------------------------------------------------------
__device__ __forceinline__ v8f energy_tile(const float* __restrict__ kb,
                                           const v2f* Bq, int lo, int hi) {
  v8f e = {};
  #pragma unroll
  for (int ks = 0; ks < 8; ++ks) {
    // A frag: A[m=q=lo, kk=d] = Kh[q][ks*4 + 2hi .. +1]
    v2f Ak = *(const v2f*)(kb + lo * HD + ks * 4 + 2 * hi);
    e = wmma4(Ak, Bq[ks], e);
  }
  return e;
}

__global__ void attn_kernel(const float* __restrict__ Qh,
                            const float* __restrict__ Kh,
                            const float* __restrict__ Vh,
                            float* __restrict__ Aout,
                            float* __restrict__ OutH) {
  __shared__ float sh_all[4 * 16 * 17];          // wave-private padded tiles
  const int lane = threadIdx.x & 31;
  const int lo = lane & 15, hi = lane >> 4;
  const int wave = threadIdx.x >> 5;
  float* sh = sh_all + wave * (16 * 17);

  const int g  = blockIdx.x * 4 + wave;          // 1024 waves total
  const int b  = g >> 9;
  const int n  = (g >> 6) & 7;
  const int p0 = (g & 63) << 4;
  const float scale = 0.0625f;                   // 1/sqrt(256)

  // Preload B fragments of Qh (16 p's x full hd=32), kept in 16 VGPRs.
  const float* qb = Qh + ((size_t)(b * NH + n) * PDIM + p0) * HD;
  v2f Bq[8];
  #pragma unroll
  for (int ks = 0; ks < 8; ++ks)
    Bq[ks] = *(const v2f*)(qb + lo * HD + ks * 4 + 2 * hi);

  const float* kb0 = Kh + (size_t)(b * NH + n) * QDIM * HD;
  const float* vb0 = Vh + (size_t)(b * NH + n) * QDIM * HD;

  // -------- pass 1: online max / sum-exp over all q --------
  float m = -1e30f, ssum = 0.0f;
  for (int qt = 0; qt < QDIM / 16; ++qt) {
    v8f e = energy_tile(kb0 + (size_t)qt * 16 * HD, Bq, lo, hi);
    float es[8];
    #pragma unroll
    for (int r = 0; r < 8; ++r) es[r] = e[r] * scale;
    float mt = es[0];
    #pragma unroll
    for (int r = 1; r < 8; ++r) mt = fmaxf(mt, es[r]);
    float mn = fmaxf(m, mt);
    float ts = 0.0f;
    #pragma unroll
    for (int r = 0; r < 8; ++r) ts += __expf(es[r] - mn);
    ssum = ssum * __expf(m - mn) + ts;
    m = mn;
  }
  // merge the two lane-halves (lane L and L+16 track the same p column)
  float m2 = __shfl_xor(m, 16, 32);
  float s2 = __shfl_xor(ssum, 16, 32);
  float M  = fmaxf(m, m2);
  float inv = 1.0f / (ssum * __expf(m - M) + s2 * __expf(m2 - M));

  // -------- pass 2: recompute, emit A, accumulate A*V --------
  v8f oa = {}, ob = {};                          // out tile: 16p x 32d
  const size_t abase = (((size_t)b * PDIM + p0 + lo) * QDIM) * NH + n;
  for (int qt = 0; qt < QDIM / 16; ++qt) {
    v8f e = energy_tile(kb0 + (size_t)qt * 16 * HD, Bq, lo, hi);
    #pragma unroll
    for (int r = 0; r < 8; ++r) {
      float av = __expf(e[r] * scale - M) * inv;
      int q = qt * 16 + r + 8 * hi;
      Aout[abase + (size_t)q * NH] = av;         // A[b][p][q][n]
      sh[(r + 8 * hi) * 17 + lo] = av;           // stage (q_local, p_local)
    }
    // A*V: D(p x d) = Aw(p x 16q) * V(16q x d); A frag read back from LDS.
    const float* vb = vb0 + (size_t)qt * 16 * HD;
    #pragma unroll
    for (int kk = 0; kk < 4; ++kk) {
      int ql = kk * 4 + 2 * hi;
      v2f Aa = { sh[ql * 17 + lo], sh[(ql + 1) * 17 + lo] };
      v2f B0 = { vb[ql * HD + lo],      vb[(ql + 1) * HD + lo]      };
      v2f B1 = { vb[ql * HD + 16 + lo], vb[(ql + 1) * HD + 16 + lo] };
      oa = wmma4(Aa, B0, oa);
      ob = wmma4(Aa, B1, ob);
    }
  }
  // store out_h as [b][p][c] with c = n*32 + d (GEMM-friendly for Wo)
  #pragma unroll
  for (int r = 0; r < 8; ++r) {
    int p = p0 + r + 8 * hi;
    float* orow = OutH + ((size_t)b * PDIM + p) * CDIM + n * HD;
    orow[lo]      = oa[r];
    orow[16 + lo] = ob[r];
  }
}

// ---------------------------------------------------------------------------
// Output projection: out[b][o][p] = sum_c Wo[o][c] * out_h[b][p][c] + bo[o]
// X layout [s][c] -> contiguous v2f B fragments.
// ---------------------------------------------------------------------------
__global__ void outproj_kernel(const float* __restrict__ W,
                               const float* __restrict__ bias,
                               const float* __restrict__ Xsc,
                               float* __restrict__ Out) {
  const int lane = threadIdx.x & 31;
  const int lo = lane & 15, hi = lane >> 4;
  const int wave = threadIdx.x >> 5;
  const int b  = blockIdx.z;
  const int m0 = blockIdx.y * 16;
  const int s0 = (blockIdx.x * 4 + wave) * 64;
  const float* Xb = Xsc + (size_t)b * PDIM * CDIM;

  v8f acc0 = {}, acc1 = {}, acc2 = {}, acc3 = {};
  #pragma unroll 4
  for (int k = 0; k < CDIM; k += 4) {
    v2f A  = *(const v2f*)(W + (size_t)(m0 + lo) * CDIM + k + 2 * hi);
    v2f B0 = *(const v2f*)(Xb + (size_t)(s0 +  0 + lo) * CDIM + k + 2 * hi);
    v2f B1 = *(const v2f*)(Xb + (size_t)(s0 + 16 + lo) * CDIM + k + 2 * hi);
    v2f B2 = *(const v2f*)(Xb + (size_t)(s0 + 32 + lo) * CDIM + k + 2 * hi);
    v2f B3 = *(const v2f*)(Xb + (size_t)(s0 + 48 + lo) * CDIM + k + 2 * hi);
    acc0 = wmma4(A, B0, acc0);
    acc1 = wmma4(A, B1, acc1);
    acc2 = wmma4(A, B2, acc2);
    acc3 = wmma4(A, B3, acc3);
  }
  #pragma unroll
  for (int r = 0; r < 8; ++r) {
    int o = m0 + r + 8 * hi;
    float bb = bias[o];
    size_t obase = ((size_t)b * CDIM + o) * PDIM;
    Out[obase + s0 +  0 + lo] = acc0[r] + bb;
    Out[obase + s0 + 16 + lo] = acc1[r] + bb;
    Out[obase + s0 + 32 + lo] = acc2[r] + bb;
    Out[obase + s0 + 48 + lo] = acc3[r] + bb;
  }
}

extern "C" void kernel_launch(void* const* d_in, const int* in_sizes, int n_in,
                              void* d_out, int out_size, void* d_ws, size_t ws_size,
                              hipStream_t stream) {
  const float* query = (const float*)d_in[0];
  const float* key   = (const float*)d_in[1];
  const float* value = (const float*)d_in[2];
  const float* Wq = (const float*)d_in[3];
  const float* bq = (const float*)d_in[4];
  const float* Wk = (const float*)d_in[5];
  const float* bk = (const float*)d_in[6];
  const float* Wv = (const float*)d_in[7];
  const float* bv = (const float*)d_in[8];
  const float* Wo = (const float*)d_in[9];
  const float* bo = (const float*)d_in[10];

  float* ws   = (float*)d_ws;
  float* Qh   = ws;                                   //  2 MB
  float* Kh   = Qh + (size_t)2 * NH * PDIM * HD;      // 16 MB
  float* Vh   = Kh + (size_t)2 * NH * QDIM * HD;      // 16 MB
  float* OutH = Vh + (size_t)2 * NH * QDIM * HD;      //  2 MB

  float* out  = (float*)d_out;                        // [B][256][32][32]
  float* Aout = out + (size_t)2 * CDIM * PDIM;        // [B][P][Q][nh]

  dim3 blk(128);  // 4 waves (wave32)
  proj_cs_kernel<<<dim3(PDIM / 256, 16, 2), blk, 0, stream>>>(Wq, bq, query, Qh, PDIM);
  proj_cs_kernel<<<dim3(QDIM / 256, 16, 2), blk, 0, stream>>>(Wk, bk, key,   Kh, QDIM);
  proj_cs_kernel<<<dim3(QDIM / 256, 16, 2), blk, 0, stream>>>(Wv, bv, value, Vh, QDIM);
  attn_kernel<<<dim3(256), blk, 0, stream>>>(Qh, Kh, Vh, Aout, OutH);
  outproj_kernel<<<dim3(PDIM / 256, 16, 2), blk, 0, stream>>>(Wo, bo, OutH, out);
}